// RNNLM_41850161333038
// MI455X (gfx1250) — compile-verified
//
#include <hip/hip_runtime.h>

// Problem sizes
#define VOCAB 32000
#define SEQ   64
#define BATCH 32
#define EMB   32
#define HID   16
#define NT16  (VOCAB / 16)        // 2000 N-tiles of 16 columns
#define TPC   25                  // tiles per chunk (400 cols)
#define NCHUNK (NT16 / TPC)       // 80 chunks per row
#define ROWS  (SEQ * BATCH)       // 2048 logits rows
#define MUNITS (SEQ * 2)          // 128 (t, m-tile) units
#define WPB   8                   // waves per block (share one N-chunk -> WGP$ reuse)

typedef float v2f __attribute__((ext_vector_type(2)));
typedef float v8f __attribute__((ext_vector_type(8)));

// ---------------------------------------------------------------------------
// Kernel 1: parallel embedding gather  X[t][b][k] = lookup[tok[t][b]][k]
// ---------------------------------------------------------------------------
__global__ void gather_x_kernel(const int* __restrict__ tok,
                                const float* __restrict__ lookup,
                                float* __restrict__ Xg) {
    int i = blockIdx.x * blockDim.x + threadIdx.x;   // 0 .. 65535
    int t = i >> 10;                                 // / (BATCH*EMB)
    int b = (i >> 5) & (BATCH - 1);
    int k = i & (EMB - 1);
    Xg[i] = lookup[(long)tok[t * BATCH + b] * EMB + k];
}

// ---------------------------------------------------------------------------
// Kernel 2: sequential Elman recurrence. One block of 512 threads: thread
// (b,j) owns h[b][j]. Weights + hidden in LDS; x double-buffered so the
// next step's embedding loads overlap the current step's FMAs + tanh.
// ---------------------------------------------------------------------------
__global__ __launch_bounds__(512) void rnn_cell_kernel(
        const float* __restrict__ Xg,
        const float* __restrict__ Wx, const float* __restrict__ Wh,
        const float* __restrict__ bx, const float* __restrict__ bh,
        const float* __restrict__ Hv,
        float* __restrict__ hAll) {
    __shared__ float xs[2][BATCH * EMB];   // double-buffered x_t
    __shared__ float hp[BATCH * HID];      // h_{t-1}
    __shared__ float swx[EMB * HID];
    __shared__ float swh[HID * HID];
    __shared__ float sb[HID];

    const int tid = threadIdx.x;           // 0..511
    const int b = tid >> 4;
    const int j = tid & 15;

    swx[tid] = Wx[tid];                    // EMB*HID == 512 exactly
    if (tid < HID * HID) swh[tid] = Wh[tid];
    if (tid < HID)       sb[tid]  = bx[tid] + bh[tid];
    hp[tid] = Hv[j];
    xs[0][tid]       = Xg[tid];
    xs[0][tid + 512] = Xg[tid + 512];
    __syncthreads();

    for (int t = 0; t < SEQ; ++t) {
        const int cur = t & 1, nxt = cur ^ 1;
        float p0 = 0.f, p1 = 0.f;
        if (t + 1 < SEQ) {                 // prefetch next step's x into regs
            p0 = Xg[(t + 1) * (BATCH * EMB) + tid];
            p1 = Xg[(t + 1) * (BATCH * EMB) + tid + 512];
        }
        float acc = sb[j];
        const float* xrow = &xs[cur][b * EMB];
        #pragma unroll
        for (int k = 0; k < EMB; ++k) acc = fmaf(xrow[k], swx[k * HID + j], acc);
        const float* hrow = &hp[b * HID];
        #pragma unroll
        for (int k = 0; k < HID; ++k) acc = fmaf(hrow[k], swh[k * HID + j], acc);
        float hv = tanhf(acc);
        __syncthreads();                   // everyone done reading hp
        hp[tid] = hv;
        hAll[t * (BATCH * HID) + tid] = hv;
        if (t + 1 < SEQ) {
            xs[nxt][tid]       = p0;
            xs[nxt][tid + 512] = p1;
        }
        __syncthreads();
    }
}

// ---------------------------------------------------------------------------
// Wave-unit decode: chunk is BLOCK-uniform (all WPB waves in a block share
// the same Wo columns -> each B tile is fetched once into WGP$ and reused
// by the other waves, cutting L2 traffic ~8x). (t, m-tile) varies per wave.
// MUNITS (=128) is divisible by WPB (=8), so chunk is constant in a block.
// ---------------------------------------------------------------------------
__device__ __forceinline__ void decode_unit(int& t, int& mtile, int& chunk) {
    int w = blockIdx.x * WPB + (threadIdx.x >> 5);   // global wave id
    chunk = w / MUNITS;                              // block-uniform
    int unit = w % MUNITS;
    t = unit >> 1;
    mtile = unit & 1;
}

// A operand (h tile, 16x16 split into four 16x4 f32 WMMA A fragments).
// A layout (ISA 7.12.2, 32-bit A 16x4): lane<16 -> K = {0,1}; lane>=16 -> {2,3}.
__device__ __forceinline__ void load_a_frag(const float* __restrict__ hAll,
                                            int t, int mtile, int half, int lm,
                                            v2f a[4]) {
    const float* hrow = hAll + (t * BATCH + mtile * 16 + lm) * HID;
    #pragma unroll
    for (int s = 0; s < 4; ++s) {
        int k0 = 4 * s + 2 * half;
        a[s].x = hrow[k0];
        a[s].y = hrow[k0 + 1];
    }
}

__device__ __forceinline__ v8f tile_logits(const v2f a[4],
                                           const float* __restrict__ Wo,
                                           int n0, int half, int lm) {
    v2f bb[4];
    #pragma unroll
    for (int s = 0; s < 4; ++s) {
        int k0 = 4 * s + 2 * half;     // B 4x16 f32: lane<16 K={0,1}, lane>=16 K={2,3}
        bb[s].x = Wo[k0 * VOCAB + n0 + lm];
        bb[s].y = Wo[(k0 + 1) * VOCAB + n0 + lm];
    }
    v8f c = {0.f, 0.f, 0.f, 0.f, 0.f, 0.f, 0.f, 0.f};
    #pragma unroll
    for (int s = 0; s < 4; ++s)
        c = __builtin_amdgcn_wmma_f32_16x16x4_f32(
                false, a[s], false, bb[s], (short)0, c, false, false);
    return c;
}

// ---------------------------------------------------------------------------
// Kernel 3: per-(row, chunk) partial sum of exp(logit). Logits bounded by
// |h|*|Wo|*K < ~5, so raw exp is numerically safe; reduction is atomic-free
// and deterministic. One wave per (t, m-tile, chunk) unit.
// ---------------------------------------------------------------------------
__global__ __launch_bounds__(WPB * 32) void logits_stats_kernel(
        const float* __restrict__ hAll, const float* __restrict__ Wo,
        float* __restrict__ partial) {
    const int lane = threadIdx.x & 31;
    const int half = lane >> 4, lm = lane & 15;
    int t, mtile, chunk;
    decode_unit(t, mtile, chunk);

    v2f a[4];
    load_a_frag(hAll, t, mtile, half, lm, a);

    float ssum[8];
    #pragma unroll
    for (int r = 0; r < 8; ++r) ssum[r] = 0.f;

    const int n0b = chunk * (TPC * 16);
    for (int tile = 0; tile < TPC; ++tile) {
        v8f c = tile_logits(a, Wo, n0b + tile * 16, half, lm);
        #pragma unroll
        for (int r = 0; r < 8; ++r) ssum[r] += __expf(c[r]);
    }
    // reduce across the 16 lanes of each half (xor masks stay within halves)
    #pragma unroll
    for (int m = 1; m < 16; m <<= 1) {
        #pragma unroll
        for (int r = 0; r < 8; ++r) ssum[r] += __shfl_xor(ssum[r], m, 32);
    }
    if (lm == 0) {   // lanes 0 and 16: rows M = r and r+8 of this tile
        const int rowBase = t * BATCH + mtile * 16 + half * 8;
        #pragma unroll
        for (int r = 0; r < 8; ++r)
            partial[(rowBase + r) * NCHUNK + chunk] = ssum[r];
    }
}

// ---------------------------------------------------------------------------
// Kernel 4: deterministic per-row reduce -> logZ[row] = log(sum exp(logits))
// ---------------------------------------------------------------------------
__global__ void reduce_logz_kernel(const float* __restrict__ partial,
                                   float* __restrict__ logZ) {
    int row = blockIdx.x * blockDim.x + threadIdx.x;
    if (row < ROWS) {
        float s = 0.f;
        #pragma unroll 4
        for (int c = 0; c < NCHUNK; ++c) s += partial[row * NCHUNK + c];
        logZ[row] = logf(s);
    }
}

// ---------------------------------------------------------------------------
// Kernel 5: recompute logits via WMMA (K=16 matmul is nearly free; Wo hits
// WGP$/L2) and stream out log_softmax = logit - logZ. Single 262 MB write.
// ---------------------------------------------------------------------------
__global__ __launch_bounds__(WPB * 32) void logits_out_kernel(
        const float* __restrict__ hAll, const float* __restrict__ Wo,
        const float* __restrict__ logZ, float* __restrict__ out) {
    const int lane = threadIdx.x & 31;
    const int half = lane >> 4, lm = lane & 15;
    int t, mtile, chunk;
    decode_unit(t, mtile, chunk);

    v2f a[4];
    load_a_frag(hAll, t, mtile, half, lm, a);

    const int rowBase = t * BATCH + mtile * 16 + half * 8;
    float lz[8];
    #pragma unroll
    for (int r = 0; r < 8; ++r) lz[r] = logZ[rowBase + r];

    const int n0b = chunk * (TPC * 16);
    for (int tile = 0; tile < TPC; ++tile) {
        const int n0 = n0b + tile * 16;
        v8f c = tile_logits(a, Wo, n0, half, lm);
        #pragma unroll
        for (int r = 0; r < 8; ++r)
            out[(long)(rowBase + r) * VOCAB + n0 + lm] = c[r] - lz[r];
    }
}

// ---------------------------------------------------------------------------
extern "C" void kernel_launch(void* const* d_in, const int* in_sizes, int n_in,
                              void* d_out, int out_size, void* d_ws, size_t ws_size,
                              hipStream_t stream) {
    (void)in_sizes; (void)n_in; (void)out_size; (void)ws_size;
    const int*   tok    = (const int*)d_in[0];
    const float* lookup = (const float*)d_in[1];
    const float* Wx     = (const float*)d_in[2];
    const float* Wh     = (const float*)d_in[3];
    const float* Wo     = (const float*)d_in[4];
    const float* bx     = (const float*)d_in[5];
    const float* bh     = (const float*)d_in[6];
    // d_in[7] = bias_o: intentionally unused (reference forward omits it)
    const float* Hv     = (const float*)d_in[8];
    float* outp = (float*)d_out;

    // workspace layout (floats): X gather | hAll | partial sums | logZ
    float* ws      = (float*)d_ws;
    float* Xg      = ws;                                   // 64*32*32 = 65536
    float* hAll    = Xg + SEQ * BATCH * EMB;               // 64*32*16 = 32768
    float* partial = hAll + SEQ * BATCH * HID;             // 2048*80  = 163840
    float* logZ    = partial + ROWS * NCHUNK;              // 2048

    gather_x_kernel<<<(SEQ * BATCH * EMB) / 256, 256, 0, stream>>>(tok, lookup, Xg);
    rnn_cell_kernel<<<1, 512, 0, stream>>>(Xg, Wx, Wh, bx, bh, Hv, hAll);

    const int units  = MUNITS * NCHUNK;    // 10240 wave-units
    const int blocks = units / WPB;        // 1280 blocks, 8 waves each
    logits_stats_kernel<<<blocks, WPB * 32, 0, stream>>>(hAll, Wo, partial);
    reduce_logz_kernel<<<(ROWS + 255) / 256, 256, 0, stream>>>(partial, logZ);
    logits_out_kernel<<<blocks, WPB * 32, 0, stream>>>(hAll, Wo, logZ, outp);
}